// Controller_88880053223466
// MI455X (gfx1250) — compile-verified
//
#include <hip/hip_runtime.h>
#include <hip/hip_bf16.h>
#include <math.h>

#define T_STACK 200000
#define V_DIM   130
#define H_DIM   512

typedef __attribute__((ext_vector_type(2))) float v2f;
typedef __attribute__((ext_vector_type(4))) float v4f;
typedef __attribute__((ext_vector_type(8))) float v8f;

__device__ __forceinline__ float sigm(float x) { return 1.0f / (1.0f + expf(-x)); }

// ---------------------------------------------------------------------------
// Per-wave fp32 GEMV tile using V_WMMA_F32_16X16X4_F32.
//   y[row0+n] += sum_k W[row0+n][k] * x[k]   for n in 0..15, K multiple of 4.
// A (16x4) holds x replicated across all 16 M rows:
//   lanes 0-15 : VGPR0=K0, VGPR1=K1 ; lanes 16-31 : VGPR0=K2, VGPR1=K3
// B (4x16) holds the 16 weight rows (n = lane&15), same K split.
// Both per-lane pairs are CONSECUTIVE floats -> single b64 loads.
// D[m][n] is identical for all m; read D[0][n] = c[0] on lanes 0-15.
// All waves of the block must call this uniformly (EXEC all ones).
// ---------------------------------------------------------------------------
__device__ __forceinline__ void gemv_wmma_add(const float* __restrict__ W, int ldw, int K,
                                              const float* __restrict__ x,  // LDS, 8B aligned
                                              float* __restrict__ y,        // LDS (+=)
                                              int rows, int wave, int nWaves, int lane)
{
    const int half2 = (lane >> 4) << 1;          // 0 or 2
    const int nl    = lane & 15;
    const float* xh = x + half2;
    const int nTiles = rows >> 4;
    for (int tile = wave; tile < nTiles; tile += nWaves) {
        const int row0 = tile << 4;
        const float* wr = W + (size_t)(row0 + nl) * (size_t)ldw + half2;
        v8f c = {0.f, 0.f, 0.f, 0.f, 0.f, 0.f, 0.f, 0.f};
        for (int kk = 0; kk < K; kk += 4) {
            v2f a = *(const v2f*)(xh + kk);                            // ds_load_b64
            v2f b = __builtin_nontemporal_load((const v2f*)(wr + kk)); // global_load_b64 NT
            c = __builtin_amdgcn_wmma_f32_16x16x4_f32(false, a, false, b,
                                                      (short)0, c, false, false);
        }
        if (lane < 16) y[row0 + lane] += c[0];
    }
}

// Fused dual GEMV (same K, same output rows): y += Wa@xa + Wb@xb
__device__ __forceinline__ void gemv_wmma_add2(const float* __restrict__ Wa,
                                               const float* __restrict__ xa,
                                               const float* __restrict__ Wb,
                                               const float* __restrict__ xb,
                                               int ldw, int K,
                                               float* __restrict__ y,
                                               int rows, int wave, int nWaves, int lane)
{
    const int half2 = (lane >> 4) << 1;
    const int nl    = lane & 15;
    const float* xah = xa + half2;
    const float* xbh = xb + half2;
    const int nTiles = rows >> 4;
    for (int tile = wave; tile < nTiles; tile += nWaves) {
        const int row0 = tile << 4;
        const float* wra = Wa + (size_t)(row0 + nl) * (size_t)ldw + half2;
        const float* wrb = Wb + (size_t)(row0 + nl) * (size_t)ldw + half2;
        v8f ca = {0.f, 0.f, 0.f, 0.f, 0.f, 0.f, 0.f, 0.f};
        v8f cb = {0.f, 0.f, 0.f, 0.f, 0.f, 0.f, 0.f, 0.f};
        for (int kk = 0; kk < K; kk += 4) {
            v2f a0 = *(const v2f*)(xah + kk);
            v2f b0 = __builtin_nontemporal_load((const v2f*)(wra + kk));
            ca = __builtin_amdgcn_wmma_f32_16x16x4_f32(false, a0, false, b0,
                                                       (short)0, ca, false, false);
            v2f a1 = *(const v2f*)(xbh + kk);
            v2f b1 = __builtin_nontemporal_load((const v2f*)(wrb + kk));
            cb = __builtin_amdgcn_wmma_f32_16x16x4_f32(false, a1, false, b1,
                                                       (short)0, cb, false, false);
        }
        if (lane < 16) y[row0 + lane] += ca[0] + cb[0];
    }
}

// ---------------------------------------------------------------------------
// K1: stream Val (200000x130 fp32, /4 exact) and the pop-scan default
//     stg[i] = max(0, prev_stg[i])  (exact once the pop carry goes <= 0).
//     One-shot stream -> non-temporal hints (native ext-vector v4f).
// ---------------------------------------------------------------------------
__global__ void copy_kernel(const float* __restrict__ prev_Val,
                            const float* __restrict__ prev_stg,
                            float* __restrict__ out_val,
                            float* __restrict__ out_stg)
{
    const size_t N4 = (size_t)T_STACK * V_DIM / 4;
    size_t i = (size_t)blockIdx.x * blockDim.x + threadIdx.x;
    const size_t stride = (size_t)gridDim.x * blockDim.x;
    const v4f* __restrict__ src = (const v4f*)prev_Val;
    v4f* __restrict__ dst = (v4f*)out_val;
    for (size_t j = i; j < N4; j += stride) {
        v4f v = __builtin_nontemporal_load(src + j);
        __builtin_nontemporal_store(v, dst + j);
    }
    for (size_t j = i; j < (size_t)T_STACK; j += stride)
        out_stg[j] = fmaxf(0.f, prev_stg[j]);
}

// ---------------------------------------------------------------------------
// K2: full controller step (one workgroup, 16 waves, WMMA GEMVs).
// ---------------------------------------------------------------------------
__global__ __launch_bounds__(512)
void controller_kernel(const float* __restrict__ input,
                       const float* __restrict__ prev_read,
                       const float* __restrict__ prev_h,
                       const float* __restrict__ prev_c,
                       const float* __restrict__ Wih0, const float* __restrict__ Whh0,
                       const float* __restrict__ bih0, const float* __restrict__ bhh0,
                       const float* __restrict__ Wih1, const float* __restrict__ Whh1,
                       const float* __restrict__ bih1, const float* __restrict__ bhh1,
                       const float* __restrict__ Wd, const float* __restrict__ Bd,
                       const float* __restrict__ Wu, const float* __restrict__ Bu,
                       const float* __restrict__ Wv, const float* __restrict__ Bv,
                       const float* __restrict__ Wo, const float* __restrict__ Bo,
                       float* __restrict__ out_ot, float* __restrict__ out_val,
                       float* __restrict__ out_stg,
                       float* __restrict__ out_h, float* __restrict__ out_c,
                       float* __restrict__ ws_ut)
{
    __shared__ __align__(16) float sX[V_DIM + 2];
    __shared__ __align__(16) float sH0[H_DIM];
    __shared__ __align__(16) float sHp1[H_DIM];
    __shared__ __align__(16) float sG[4 * H_DIM];
    __shared__ __align__(16) float sH1[H_DIM];
    __shared__ __align__(16) float sH2[H_DIM];

    const int t    = threadIdx.x;
    const int lane = t & 31;
    const int wave = t >> 5;
    const int nw   = blockDim.x >> 5;

    if (t < V_DIM) sX[t] = input[t] + prev_read[t];
    if (t < H_DIM) { sH0[t] = prev_h[t]; sHp1[t] = prev_h[H_DIM + t]; }
    __syncthreads();

    // ---- layer 0: biases + Wih0 @ x (K=130, VALU), then += Whh0 @ h0 (WMMA)
    for (int r = t; r < 4 * H_DIM; r += blockDim.x) {
        float acc = bih0[r] + bhh0[r];
        const float* w = Wih0 + (size_t)r * V_DIM;
        for (int k = 0; k < V_DIM; ++k) acc += w[k] * sX[k];
        sG[r] = acc;
    }
    __syncthreads();
    gemv_wmma_add(Whh0, H_DIM, H_DIM, sH0, sG, 4 * H_DIM, wave, nw, lane);
    __syncthreads();

    // ---- gates layer 0 (pytorch order i,f,g,o)
    if (t < H_DIM) {
        float ig = sigm(sG[t]);
        float fg = sigm(sG[H_DIM + t]);
        float gg = tanhf(sG[2 * H_DIM + t]);
        float og = sigm(sG[3 * H_DIM + t]);
        float c1 = fg * prev_c[t] + ig * gg;
        float h1 = og * tanhf(c1);
        sH1[t] = h1;
        out_h[t] = h1;
        out_c[t] = c1;
    }
    __syncthreads();

    // ---- layer 1: biases, then += Wih1 @ h1 + Whh1 @ h_prev1 (fused WMMA)
    for (int r = t; r < 4 * H_DIM; r += blockDim.x) sG[r] = bih1[r] + bhh1[r];
    __syncthreads();
    gemv_wmma_add2(Wih1, sH1, Whh1, sHp1, H_DIM, H_DIM, sG, 4 * H_DIM, wave, nw, lane);
    __syncthreads();

    // ---- gates layer 1
    if (t < H_DIM) {
        float ig = sigm(sG[t]);
        float fg = sigm(sG[H_DIM + t]);
        float gg = tanhf(sG[2 * H_DIM + t]);
        float og = sigm(sG[3 * H_DIM + t]);
        float c2 = fg * prev_c[H_DIM + t] + ig * gg;
        float h2 = og * tanhf(c2);
        sH2[t] = h2;
        out_h[H_DIM + t] = h2;
        out_c[H_DIM + t] = c2;
    }
    __syncthreads();

    // ---- dt / ut scalar heads (partials in sG, serial reduce)
    if (t < H_DIM) { sG[t] = sH2[t] * Wd[t]; sG[H_DIM + t] = sH2[t] * Wu[t]; }
    __syncthreads();
    if (t == 0) {
        float sd = 0.f, su = 0.f;
        for (int k = 0; k < H_DIM; ++k) { sd += sG[k]; su += sG[H_DIM + k]; }
        out_stg[T_STACK] = sigm(sd + Bd[0]);   // dt = new top strength
        ws_ut[0] = sigm(su + Bu[0]);           // ut for the pop scan
    }

    // ---- vt head -> Val row T (pushed value)
    if (t < V_DIM) {
        float acc = Bv[t];
        for (int k = 0; k < H_DIM; ++k) acc += sH2[k] * Wv[(size_t)k * V_DIM + t];
        out_val[(size_t)T_STACK * V_DIM + t] = tanhf(acc);
    }

    // ---- ot head
    if (t < H_DIM) {
        float acc = Bo[t];
        for (int k = 0; k < H_DIM; ++k) acc += sH2[k] * Wo[(size_t)k * H_DIM + t];
        out_ot[t] = tanhf(acc);
    }
}

// ---------------------------------------------------------------------------
// K3: serial pop scan from the top. Once u <= 0, the remaining outputs equal
//     the max(0, prev_stg[i]) defaults already written by K1 -> early exit.
// ---------------------------------------------------------------------------
__global__ void pop_kernel(const float* __restrict__ prev_stg,
                           float* __restrict__ out_stg,
                           const float* __restrict__ ws_ut)
{
    if (threadIdx.x != 0 || blockIdx.x != 0) return;
    float u = ws_ut[0];
    for (int i = T_STACK - 1; i >= 0; --i) {
        float s  = prev_stg[i];
        float sn = fmaxf(0.f, s - fmaxf(0.f, u));
        out_stg[i] = sn;
        u -= sn;
        if (u <= 0.f) break;
    }
}

// ---------------------------------------------------------------------------
// K4: read pass. coef_i = min(stg[i], max(0, 1 - suffix_sum)); all stg >= 0,
//     so coefs are identically 0 once the suffix sum reaches 1 -> early exit.
//     130 lanes accumulate rt; prefetch the next Val row while FMA-ing.
// ---------------------------------------------------------------------------
__global__ __launch_bounds__(160)
void read_kernel(const float* __restrict__ stg,
                 const float* __restrict__ val,
                 float* __restrict__ rt)
{
    __shared__ float sc;
    __shared__ int   sstop;
    const int t = threadIdx.x;
    float acc = 0.f;
    float S   = 0.f;   // maintained by thread 0 only
    for (int i = T_STACK; i >= 0; --i) {
        if (t == 0) {
            float s = stg[i];
            sc = fminf(s, fmaxf(0.f, 1.f - S));
            S += s;
            sstop = (S >= 1.f) ? 1 : 0;
        }
        __syncthreads();
        const float coef = sc;
        const int   stop = sstop;
        if (t < V_DIM) {
            if (coef != 0.f) acc += coef * val[(size_t)i * V_DIM + t];
            if (!stop && i > 0)
                __builtin_prefetch(&val[(size_t)(i - 1) * V_DIM + t], 0, 0);
        }
        __syncthreads();
        if (stop) break;
    }
    if (t < V_DIM) rt[t] = acc;
}

// ---------------------------------------------------------------------------
extern "C" void kernel_launch(void* const* d_in, const int* in_sizes, int n_in,
                              void* d_out, int out_size, void* d_ws, size_t ws_size,
                              hipStream_t stream)
{
    const float* input     = (const float*)d_in[0];
    const float* prev_Val  = (const float*)d_in[1];
    const float* prev_stg  = (const float*)d_in[2];
    const float* prev_read = (const float*)d_in[3];
    const float* prev_h    = (const float*)d_in[4];
    const float* prev_c    = (const float*)d_in[5];
    const float* Wih0      = (const float*)d_in[6];
    const float* Whh0      = (const float*)d_in[7];
    const float* bih0      = (const float*)d_in[8];
    const float* bhh0      = (const float*)d_in[9];
    const float* Wih1      = (const float*)d_in[10];
    const float* Whh1      = (const float*)d_in[11];
    const float* bih1      = (const float*)d_in[12];
    const float* bhh1      = (const float*)d_in[13];
    const float* Wd        = (const float*)d_in[14];
    const float* Bd        = (const float*)d_in[15];
    const float* Wu        = (const float*)d_in[16];
    const float* Bu        = (const float*)d_in[17];
    const float* Wv        = (const float*)d_in[18];
    const float* Bv        = (const float*)d_in[19];
    const float* Wo        = (const float*)d_in[20];
    const float* Bo        = (const float*)d_in[21];

    // Output layout (flat, return order): ot | Val | stg | rt | h_new | c_new
    float* out     = (float*)d_out;
    float* out_ot  = out;
    float* out_val = out_ot + H_DIM;
    float* out_stg = out_val + (size_t)(T_STACK + 1) * V_DIM;
    float* out_rt  = out_stg + (T_STACK + 1);
    float* out_h   = out_rt + V_DIM;
    float* out_c   = out_h + 2 * H_DIM;

    float* ws_ut = (float*)d_ws;

    // K1: bulk Val copy + pop-scan defaults (memory-bound, ~104MB r + 104MB w)
    copy_kernel<<<2048, 256, 0, stream>>>(prev_Val, prev_stg, out_val, out_stg);

    // K2: controller step (WMMA fp32 GEMVs)
    controller_kernel<<<1, 512, 0, stream>>>(input, prev_read, prev_h, prev_c,
                                             Wih0, Whh0, bih0, bhh0,
                                             Wih1, Whh1, bih1, bhh1,
                                             Wd, Bd, Wu, Bu, Wv, Bv, Wo, Bo,
                                             out_ot, out_val, out_stg,
                                             out_h, out_c, ws_ut);

    // K3: pop scan (serial walker with exact early exit)
    pop_kernel<<<1, 32, 0, stream>>>(prev_stg, out_stg, ws_ut);

    // K4: read pass -> rt
    read_kernel<<<1, 160, 0, stream>>>(out_stg, out_val, out_rt);
}